// MIX_56478819943005
// MI455X (gfx1250) — compile-verified
//
#include <hip/hip_runtime.h>
#include <hip/hip_bf16.h>

typedef __attribute__((ext_vector_type(2))) float v2f;
typedef __attribute__((ext_vector_type(8))) float v8f;

#define TPB 256

// ---------------- degree / normalization ----------------
__global__ void k_init_deg(float* deg, int N) {
  int i = blockIdx.x * blockDim.x + threadIdx.x;
  if (i < N) deg[i] = 1.0f;  // self-loop
}

__global__ void k_edge_deg(const int* __restrict__ dst, float* deg, int E) {
  int e = blockIdx.x * blockDim.x + threadIdx.x;
  if (e < E) atomicAdd(&deg[dst[e]], 1.0f);
}

__global__ void k_rsqrt(const float* __restrict__ deg, float* dinv, int N) {
  int i = blockIdx.x * blockDim.x + threadIdx.x;
  if (i < N) dinv[i] = rsqrtf(deg[i]);
}

// ---------------- fp32 WMMA GEMM: C[Nrows x OUT] = A[Nrows x K] @ B[K x OUT] --------
// One wave computes a 32x32 block = 2x2 tiles of 16x16, V_WMMA_F32_16X16X4_F32.
// A fragments reused across the 2 column tiles, B fragments across the 2 row tiles:
// per K-step (K+=4): 2x global_load_b64 + 4x global_load_b32 feed 4 WMMAs.
// Shared device body; K/OUT may be compile-time constants (templated instantiations
// below) so B-side addresses fold into 24-bit instruction offsets.
template <int KC, int OUTC>
__device__ __forceinline__ void gemm_body(const float* __restrict__ A,
                                          const float* __restrict__ B,
                                          float* __restrict__ C,
                                          int Nrows, int Kr, int OUTr) {
  const int K   = (KC   > 0) ? KC   : Kr;
  const int OUT = (OUTC > 0) ? OUTC : OUTr;
  int gtid = blockIdx.x * blockDim.x + threadIdx.x;
  int wave = gtid >> 5;
  int lane = threadIdx.x & 31;
  int rowTiles = Nrows >> 4;                 // 16-row tiles (3125 for N=50000)
  int rowB = (rowTiles + 1) >> 1;            // 32-row blocks (ceil)
  int colB = OUT >> 5;                       // 32-col blocks (OUT % 32 == 0)
  if (wave >= rowB * colB) return;           // wave-uniform exit: EXEC all-1 at WMMA
  int rb = wave / colB;
  int cb = wave % colB;
  int row0 = rb << 5;
  int col0 = cb << 5;
  int half = lane >> 4;                      // 0: lanes 0-15, 1: lanes 16-31
  int l    = lane & 15;

  // second row tile: clamp loads if it falls off the end; skip its store below
  int row1    = row0 + 16;
  int row1_ok = (row1 + 16 <= Nrows);
  int row1_ld = row1_ok ? row1 : (Nrows - 16);

  const float* arow0 = A + (size_t)(row0 + l) * K + 2 * half;
  const float* arow1 = A + (size_t)(row1_ld + l) * K + 2 * half;
  const float* bbase = B + (size_t)(2 * half) * OUT + col0 + l;

  v8f acc00 = {}, acc01 = {}, acc10 = {}, acc11 = {};
#pragma unroll 8
  for (int k = 0; k < K; k += 4) {
    // A 16x4 fragments: lane(half,l) holds A[row][k+2h], A[row][k+2h+1]
    float2 av0 = *(const float2*)(arow0 + k);   // 8B aligned: K%4==0
    float2 av1 = *(const float2*)(arow1 + k);
    v2f a0; a0.x = av0.x; a0.y = av0.y;
    v2f a1; a1.x = av1.x; a1.y = av1.y;
    // B 4x16 fragments: VGPR0 = row k+2h, VGPR1 = row k+2h+1
    const float* bp = bbase + (size_t)k * OUT;
    v2f b0, b1;
    b0.x = bp[0];
    b0.y = bp[(size_t)OUT];
    b1.x = bp[16];
    b1.y = bp[(size_t)OUT + 16];
    acc00 = __builtin_amdgcn_wmma_f32_16x16x4_f32(false, a0, false, b0, (short)0, acc00,
                                                  false, false);
    acc01 = __builtin_amdgcn_wmma_f32_16x16x4_f32(false, a0, false, b1, (short)0, acc01,
                                                  false, false);
    acc10 = __builtin_amdgcn_wmma_f32_16x16x4_f32(false, a1, false, b0, (short)0, acc10,
                                                  false, false);
    acc11 = __builtin_amdgcn_wmma_f32_16x16x4_f32(false, a1, false, b1, (short)0, acc11,
                                                  false, false);
  }

  // C/D layout: VGPR r -> M = tileRow + r + 8*half, N = col + l
  float* c00 = C + (size_t)(row0 + 8 * half) * OUT + col0 + l;
  for (int r = 0; r < 8; ++r) {
    c00[(size_t)r * OUT]      = acc00[r];
    c00[(size_t)r * OUT + 16] = acc01[r];
  }
  if (row1_ok) {                              // wave-uniform condition
    float* c10 = C + (size_t)(row1 + 8 * half) * OUT + col0 + l;
    for (int r = 0; r < 8; ++r) {
      c10[(size_t)r * OUT]      = acc10[r];
      c10[(size_t)r * OUT + 16] = acc11[r];
    }
  }
}

template <int KC, int OUTC>
__global__ void k_gemm_wmma_t(const float* __restrict__ A, const float* __restrict__ B,
                              float* __restrict__ C, int Nrows) {
  gemm_body<KC, OUTC>(A, B, C, Nrows, KC, OUTC);
}

__global__ void k_gemm_wmma(const float* __restrict__ A, const float* __restrict__ B,
                            float* __restrict__ C, int Nrows, int K, int OUT) {
  gemm_body<0, 0>(A, B, C, Nrows, K, OUT);
}

// ---------------- aggregation ----------------
// out[i,f] = bias[f] + hlin[i,f] * dinv[i]^2   (self-loop contribution + bias)
__global__ void k_agg_init(const float* __restrict__ hlin, const float* __restrict__ dinv,
                           const float* __restrict__ bias, float* out, int N, int OUT) {
  long long t = (long long)blockIdx.x * blockDim.x + threadIdx.x;
  if (t >= (long long)N * OUT) return;
  int i = (int)(t / OUT);
  int f = (int)(t % OUT);
  float di = dinv[i];
  out[t] = bias[f] + hlin[t] * di * di;
}

// hout[dst,f..f+3] += hin[src,f..f+3] * dinv[src]*dinv[dst]
__global__ void k_scatter(const int* __restrict__ src, const int* __restrict__ dst,
                          const float* __restrict__ dinv, const float* __restrict__ hin,
                          float* hout, int E, int OUT) {
  int v4 = OUT >> 2;
  long long t = (long long)blockIdx.x * blockDim.x + threadIdx.x;
  if (t >= (long long)E * v4) return;
  int e = (int)(t / v4);
  int f = (int)(t % v4) << 2;
  int s = src[e], d = dst[e];
  float w = dinv[s] * dinv[d];
  float4 h = *(const float4*)(hin + (size_t)s * OUT + f);
  float* o = hout + (size_t)d * OUT + f;
  atomicAdd(o + 0, h.x * w);
  atomicAdd(o + 1, h.y * w);
  atomicAdd(o + 2, h.z * w);
  atomicAdd(o + 3, h.w * w);
}

__global__ void k_lrelu(float* x, long long n) {
  long long t = (long long)blockIdx.x * blockDim.x + threadIdx.x;
  if (t < n) { float v = x[t]; x[t] = v > 0.f ? v : 0.01f * v; }
}

// ---------------- segment max pool (order-preserving uint encoding) ----------------
__device__ __forceinline__ unsigned encf(float f) {
  unsigned u = __float_as_uint(f);
  return (u & 0x80000000u) ? ~u : (u | 0x80000000u);
}
__device__ __forceinline__ float decf(unsigned u) {
  return __uint_as_float((u & 0x80000000u) ? (u & 0x7fffffffu) : ~u);
}

__global__ void k_pool_init(unsigned* enc, int n) {
  int t = blockIdx.x * blockDim.x + threadIdx.x;
  if (t < n) enc[t] = 0x007FFFFFu;  // enc(-inf)
}

__global__ void k_pool_max(const float* __restrict__ h, const int* __restrict__ batch,
                           unsigned* enc, int N, int OUT) {
  long long t = (long long)blockIdx.x * blockDim.x + threadIdx.x;
  if (t >= (long long)N * OUT) return;
  int i = (int)(t / OUT);
  int f = (int)(t % OUT);
  atomicMax(&enc[(size_t)batch[i] * OUT + f], encf(h[t]));
}

__global__ void k_pool_dec(const unsigned* __restrict__ enc, float* pooled, int n) {
  int t = blockIdx.x * blockDim.x + threadIdx.x;
  if (t < n) pooled[t] = decf(enc[t]);
}

// ---------------- MLP head (tiny: G=50 rows) ----------------
// z1[g,j] = relu(ba[j] + sum_k pooled[g,k] * (Wa[k,j] + Wa[k+Kc,j]))   (concat fold)
__global__ void k_mlp_concat(const float* __restrict__ pooled, const float* __restrict__ Wa,
                             const float* __restrict__ ba, float* z, int G, int Kc, int OUT) {
  int t = blockIdx.x * blockDim.x + threadIdx.x;
  if (t >= G * OUT) return;
  int g = t / OUT, j = t % OUT;
  const float* p = pooled + (size_t)g * Kc;
  float acc = ba[j];
  for (int k = 0; k < Kc; ++k)
    acc += p[k] * (Wa[(size_t)k * OUT + j] + Wa[(size_t)(k + Kc) * OUT + j]);
  z[t] = fmaxf(acc, 0.f);
}

__global__ void k_dense(const float* __restrict__ in, const float* __restrict__ W,
                        const float* __restrict__ bias, float* out,
                        int G, int K, int OUT, int relu) {
  int t = blockIdx.x * blockDim.x + threadIdx.x;
  if (t >= G * OUT) return;
  int g = t / OUT, j = t % OUT;
  const float* p = in + (size_t)g * K;
  float acc = bias[j];
  for (int k = 0; k < K; ++k) acc += p[k] * W[(size_t)k * OUT + j];
  out[t] = relu ? fmaxf(acc, 0.f) : acc;
}

// ---------------- host ----------------
static inline unsigned nblk(long long n) { return (unsigned)((n + TPB - 1) / TPB); }

extern "C" void kernel_launch(void* const* d_in, const int* in_sizes, int n_in,
                              void* d_out, int out_size, void* d_ws, size_t ws_size,
                              hipStream_t stream) {
  const float* x     = (const float*)d_in[0];
  const int*   eidx  = (const int*)d_in[1];
  const int*   batch = (const int*)d_in[2];
  const float* W1 = (const float*)d_in[3];  const float* b1 = (const float*)d_in[4];
  const float* W2 = (const float*)d_in[5];  const float* b2 = (const float*)d_in[6];
  const float* W3 = (const float*)d_in[7];  const float* b3 = (const float*)d_in[8];
  const float* Wa = (const float*)d_in[9];  const float* ba = (const float*)d_in[10];
  const float* Wb = (const float*)d_in[11]; const float* bb = (const float*)d_in[12];
  const float* Wc = (const float*)d_in[13]; const float* bc = (const float*)d_in[14];
  float* out = (float*)d_out;

  const int H  = in_sizes[4];            // 64
  const int F  = in_sizes[3] / H;        // 128
  const int N  = in_sizes[0] / F;        // 50000
  const int E  = in_sizes[1] / 2;        // 800000
  const int O2 = in_sizes[6];            // 128
  const int O3 = in_sizes[8];            // 256
  const int AO = in_sizes[10];           // 1024
  const int BO = in_sizes[12];           // 512
  const int CO = in_sizes[14];           // 4
  const int G  = out_size / CO;          // 50

  const int* src = eidx;
  const int* dst = eidx + E;

  float* ws = (float*)d_ws;
  float* deg  = ws;                         ws += N;
  float* dinv = ws;                         ws += N;
  float* bufA = ws;                         ws += (size_t)N * O3;  // GEMM outputs
  float* bufB = ws;                         ws += (size_t)N * O3;  // aggregated
  unsigned* penc = (unsigned*)ws;           ws += (size_t)G * O3;
  float* pooled = ws;                       ws += (size_t)G * O3;
  float* z1 = ws;                           ws += (size_t)G * AO;
  float* z2 = ws;                           ws += (size_t)G * BO;
  (void)ws_size; (void)n_in;

  // normalization
  k_init_deg<<<nblk(N), TPB, 0, stream>>>(deg, N);
  k_edge_deg<<<nblk(E), TPB, 0, stream>>>(dst, deg, E);
  k_rsqrt<<<nblk(N), TPB, 0, stream>>>(deg, dinv, N);

  auto gemm = [&](const float* A, const float* Bm, float* C, int K, int OUT) {
    long long rowB = ((N >> 4) + 1) >> 1;
    long long waves = rowB * (OUT >> 5);
    unsigned blocks = nblk(waves * 32);
    if (K == 128 && OUT == 64)
      k_gemm_wmma_t<128, 64><<<blocks, TPB, 0, stream>>>(A, Bm, C, N);
    else if (K == 64 && OUT == 128)
      k_gemm_wmma_t<64, 128><<<blocks, TPB, 0, stream>>>(A, Bm, C, N);
    else if (K == 128 && OUT == 256)
      k_gemm_wmma_t<128, 256><<<blocks, TPB, 0, stream>>>(A, Bm, C, N);
    else
      k_gemm_wmma<<<blocks, TPB, 0, stream>>>(A, Bm, C, N, K, OUT);
  };
  auto layer = [&](const float* hin, const float* W, const float* b, int K, int OUT,
                   int do_lrelu) {
    gemm(hin, W, bufA, K, OUT);
    k_agg_init<<<nblk((long long)N * OUT), TPB, 0, stream>>>(bufA, dinv, b, bufB, N, OUT);
    k_scatter<<<nblk((long long)E * (OUT >> 2)), TPB, 0, stream>>>(src, dst, dinv, bufA,
                                                                   bufB, E, OUT);
    if (do_lrelu)
      k_lrelu<<<nblk((long long)N * OUT), TPB, 0, stream>>>(bufB, (long long)N * OUT);
  };

  layer(x,    W1, b1, F,  H,  1);
  layer(bufB, W2, b2, H,  O2, 1);
  layer(bufB, W3, b3, O2, O3, 0);

  // global max pool over batch segments
  k_pool_init<<<nblk(G * O3), TPB, 0, stream>>>(penc, G * O3);
  k_pool_max<<<nblk((long long)N * O3), TPB, 0, stream>>>(bufB, batch, penc, N, O3);
  k_pool_dec<<<nblk(G * O3), TPB, 0, stream>>>(penc, pooled, G * O3);

  // MLP head: concat(pooled,pooled)@Wa folded into Wa_top + Wa_bot
  k_mlp_concat<<<nblk(G * AO), TPB, 0, stream>>>(pooled, Wa, ba, z1, G, O3, AO);
  k_dense<<<nblk(G * BO), TPB, 0, stream>>>(z1, Wb, bb, z2, G, AO, BO, 1);
  k_dense<<<nblk(G * CO), TPB, 0, stream>>>(z2, Wc, bc, out, G, BO, CO, 0);
}